// SAGEConv_56556129354467
// MI455X (gfx1250) — compile-verified
//
#include <hip/hip_runtime.h>

// SAGEConv (mean aggregator) for MI455X / gfx1250.
//
// Roofline: aggregation moves ~115 MB through HBM (~5us @ 23.3 TB/s) and does
// 102M f32 atomics (feat is 25.6 MB -> L2-resident gathers). The two
// 50000x128x128 GEMMs (3.3 GFLOP) are fused into one WMMA kernel using the
// exact-precision fp32 matrix path V_WMMA_F32_16X16X4_F32 (reference is fp32,
// so no precision downgrade).

typedef __attribute__((ext_vector_type(2))) float v2f;
typedef __attribute__((ext_vector_type(8))) float v8f;

#define N_NODES 50000
#define N_EDGES 800000
#define D       128
#define ROW_BLOCKS (N_NODES / 16)   // 3125, exact

// ---------------------------------------------------------------------------
// K1: zero the accumulator and degree arrays in workspace.
// ---------------------------------------------------------------------------
__global__ void sage_zero(float* __restrict__ acc, float* __restrict__ deg) {
    int idx = blockIdx.x * 256 + threadIdx.x;
    if (idx < N_NODES * D) acc[idx] = 0.0f;
    if (idx < N_NODES)     deg[idx] = 0.0f;
}

// ---------------------------------------------------------------------------
// K2: edge-parallel scatter. One thread per (edge, feature).
//   acc[dst[e], f] += feat[src[e], f];   deg[dst[e]] += 1 (lane f==0)
// feat (25.6 MB) stays L2-resident, so the random gather is cheap; atomics
// land in L2's atomic units.
// ---------------------------------------------------------------------------
__global__ void sage_scatter(const float* __restrict__ feat,
                             const int*   __restrict__ src,
                             const int*   __restrict__ dst,
                             float* __restrict__ acc,
                             float* __restrict__ deg) {
    long long idx = (long long)blockIdx.x * 256 + threadIdx.x;
    int e = (int)(idx >> 7);        // edge index
    int f = (int)(idx & (D - 1));   // feature index
    if (e >= N_EDGES) return;
    int s = src[e];
    int d = dst[e];
    float v = feat[(long long)s * D + f];
    atomicAdd(&acc[(long long)d * D + f], v);
    if (f == 0) atomicAdd(&deg[d], 1.0f);
}

// ---------------------------------------------------------------------------
// K3: fused WMMA GEMM:
//   out = feat @ Ws^T  +  (acc / max(deg,1)) @ Wn^T  +  b_self
//
// Grid: 3125 blocks x 256 threads (8 waves). Block r handles node rows
// [16r, 16r+16); wave w handles output channels [16w, 16w+16).
// fp32 WMMA fragment layout (ISA 7.12.2, 16x4 A / 4x16 B, 32-bit):
//   A: lane l<16 -> M=l,    VGPR0=K0, VGPR1=K1 ; lane>=16 -> M=l-16, K2/K3
//   B: lane l<16 -> N=l,    VGPR0=K0, VGPR1=K1 ; lane>=16 -> N=l-16, K2/K3
//      B[K][N] = W[N][K] (W is [out,in]) -> a W row is contiguous in K.
//   C: VGPR i: lane<16 -> (M=i, N=l) ; lane>=16 -> (M=i+8, N=l-16)
// EXEC is all-1s: no divergence anywhere in this kernel.
// ---------------------------------------------------------------------------
__global__ __launch_bounds__(256)
void sage_wmma_gemm(const float* __restrict__ feat,
                    const float* __restrict__ acc,
                    const float* __restrict__ deg,
                    const float* __restrict__ Wn,   // [128,128] fc_neigh
                    const float* __restrict__ Ws,   // [128,128] fc_self
                    const float* __restrict__ bs,   // [128]
                    float* __restrict__ out) {
    const int rowBlock = blockIdx.x;          // 0..3124
    const int wave = threadIdx.x >> 5;        // 0..7  -> output-channel tile
    const int lane = threadIdx.x & 31;
    const int half = lane >> 4;               // 0 or 1
    const int l15  = lane & 15;
    const int koff = half * 2;                // K sub-offset within a k-step

    const int arow = rowBlock * 16 + l15;     // node row this lane feeds (A)
    const int wrow = wave * 16 + l15;         // output channel this lane feeds (B)

    const float invdeg = 1.0f / fmaxf(deg[arow], 1.0f);

    const float* __restrict__ aS = feat + (long long)arow * D + koff; // self A
    const float* __restrict__ aN = acc  + (long long)arow * D + koff; // neigh A
    const float* __restrict__ bS = Ws   + (long long)wrow * D + koff; // self B
    const float* __restrict__ bN = Wn   + (long long)wrow * D + koff; // neigh B

    v8f c = {0.f, 0.f, 0.f, 0.f, 0.f, 0.f, 0.f, 0.f};

#pragma unroll
    for (int k = 0; k < D; k += 4) {
        // self contribution: feat @ Ws^T
        v2f a0 = *(const v2f*)(aS + k);       // 8B-aligned -> global_load_b64
        v2f b0 = *(const v2f*)(bS + k);
        c = __builtin_amdgcn_wmma_f32_16x16x4_f32(
                /*neg_a=*/false, a0, /*neg_b=*/false, b0,
                /*c_mod=*/(short)0, c, /*reuse_a=*/false, /*reuse_b=*/false);

        // neighbor contribution: (acc * invdeg) @ Wn^T
        v2f a1 = *(const v2f*)(aN + k);
        a1 = a1 * invdeg;                     // fold the mean into A
        v2f b1 = *(const v2f*)(bN + k);
        c = __builtin_amdgcn_wmma_f32_16x16x4_f32(
                false, a1, false, b1, (short)0, c, false, false);
    }

    // Write back with bias. C VGPR i -> (M = i + 8*half, N = l15) of the tile.
    const int ncol = wave * 16 + l15;
    const float bias = bs[ncol];
#pragma unroll
    for (int i = 0; i < 8; ++i) {
        int m = rowBlock * 16 + i + half * 8;
        out[(long long)m * D + ncol] = c[i] + bias;
    }
}

// ---------------------------------------------------------------------------
extern "C" void kernel_launch(void* const* d_in, const int* in_sizes, int n_in,
                              void* d_out, int out_size, void* d_ws, size_t ws_size,
                              hipStream_t stream) {
    const float* feat = (const float*)d_in[0];
    const int*   src  = (const int*)d_in[1];
    const int*   dst  = (const int*)d_in[2];
    const float* Wn   = (const float*)d_in[3];
    const float* Ws   = (const float*)d_in[4];
    const float* bs   = (const float*)d_in[5];
    float* out = (float*)d_out;

    float* acc = (float*)d_ws;                      // N_NODES*D floats (25.6 MB)
    float* deg = acc + (size_t)N_NODES * D;         // N_NODES floats

    // K1: zero workspace
    {
        int total = N_NODES * D;
        int grid = (total + 255) / 256;
        sage_zero<<<grid, 256, 0, stream>>>(acc, deg);
    }
    // K2: scatter edges (one thread per edge-feature)
    {
        long long total = (long long)N_EDGES * D;   // 102.4M threads
        int grid = (int)((total + 255) / 256);      // 400000 blocks
        sage_scatter<<<grid, 256, 0, stream>>>(feat, src, dst, acc, deg);
    }
    // K3: fused fp32 WMMA GEMM + bias
    sage_wmma_gemm<<<ROW_BLOCKS, 256, 0, stream>>>(feat, acc, deg, Wn, Ws, bs, out);
}